// PairwiseLoss_4355096838435
// MI455X (gfx1250) — compile-verified
//
#include <hip/hip_runtime.h>
#include <hip/hip_bf16.h>

// Problem constants (from reference setup_inputs)
#define HH 256
#define WW 256
#define EE 8
#define TW 32          // tile width  (one wave spans it -> coalesced, LDS conflict-free)
#define TH 32          // tile height
#define HALO 2         // dilation-2, 3x3 kernel -> halo of 2
#define LW (TW + 2*HALO)   // 36
#define LH (TH + 2*HALO)   // 36
#define LSTRIDE LW

typedef float v2f __attribute__((ext_vector_type(2)));
typedef float v8f __attribute__((ext_vector_type(8)));

// Stage 1: per-block (32x32 tile of one image) weighted pairwise-loss partial sums.
__global__ __launch_bounds__(256)
void pairwise_stage1(const float* __restrict__ ics,     // [N,8,H,W]
                     const float* __restrict__ maskp,   // [N,H,W]
                     const float* __restrict__ gtb,     // [N,H,W]
                     float2* __restrict__ partial)      // [gridDim.y*gridDim.x]
{
  __shared__ float sLF[LH * LSTRIDE];  // log_sigmoid(x)
  __shared__ float sLB[LH * LSTRIDE];  // log_sigmoid(-x)

  const int n  = blockIdx.y;
  const int tilesX = WW / TW;                    // 8
  const int h0 = (blockIdx.x / tilesX) * TH;
  const int w0 = (blockIdx.x % tilesX) * TW;
  const int tid = threadIdx.x;

  const size_t planeHW = (size_t)HH * WW;
  const float* maskN = maskp + (size_t)n * planeHW;
  const float* gtN   = gtb   + (size_t)n * planeHW;
  const float* icsN  = ics   + (size_t)n * EE * planeHW;

  // Cooperative halo-tile load + one softplus per pixel (reused by up to 9 taps).
  // OOB pixels get lf = lb = 0, which reproduces JAX's zero-pad of log values:
  // logaddexp(lf_p+0, lb_p+0) = log(sig + (1-sig)) = 0  -> tap loss 0.
  for (int i = tid; i < LH * LW; i += 256) {
    const int lh = i / LW, lw = i % LW;
    const int h = h0 + lh - HALO, w = w0 + lw - HALO;
    float lf = 0.f, lb = 0.f;
    if ((unsigned)h < (unsigned)HH && (unsigned)w < (unsigned)WW) {
      const float x = maskN[(size_t)h * WW + w];
      const float t = log1pf(__expf(-fabsf(x)));   // softplus(-|x|), overflow-safe
      lf = fminf(x, 0.f) - t;                      // log_sigmoid(x)
      lb = lf - x;                                 // log_sigmoid(-x)
    }
    sLF[lh * LSTRIDE + lw] = lf;
    sLB[lh * LSTRIDE + lw] = lb;
  }
  __syncthreads();

  const int lx  = tid & 31;   // wave spans x: contiguous 128B vmem, 0-conflict LDS
  const int ly0 = tid >> 5;   // 8 waves span y
  const int w   = w0 + lx;

  // Row-major 3x3 taps minus center, dilation 2 (matches F.unfold order).
  const int DY[8] = {-2,-2,-2, 0, 0, 2, 2, 2};
  const int DX[8] = {-2, 0, 2,-2, 2,-2, 0, 2};

  float num = 0.f, den = 0.f;

  for (int ly = ly0; ly < TH; ly += 8) {
    const int h = h0 + ly;
    if (ly + 8 < TH)  // prefetch next gt row for this wave (global_prefetch_b8)
      __builtin_prefetch(&gtN[(size_t)(h + 8) * WW + w], 0, 0);
    const float gtv = gtN[(size_t)h * WW + w];
    const int c = (ly + HALO) * LSTRIDE + (lx + HALO);
    const float lfp = sLF[c], lbp = sLB[c];
#pragma unroll
    for (int e = 0; e < 8; ++e) {
      const int q = c + DY[e] * LSTRIDE + DX[e];
      const float a = lfp + sLF[q];
      const float b = lbp + sLB[q];
      const float m = fmaxf(a, b);
      const float loss = -(m + log1pf(__expf(fminf(a, b) - m)));  // -logaddexp
      const float ic  = icsN[(size_t)e * planeHW + (size_t)h * WW + w];
      const float wgt = (ic >= 0.3f) ? gtv : 0.f;
      num += loss * wgt;
      den += wgt;
    }
  }

  // wave32 lane-tree reduce
#pragma unroll
  for (int off = 16; off > 0; off >>= 1) {
    num += __shfl_down(num, off, 32);
    den += __shfl_down(den, off, 32);
  }
  __shared__ float2 sPart[8];
  if (lx == 0) sPart[ly0] = make_float2(num, den);
  __syncthreads();
  if (tid == 0) {
    float2 acc = make_float2(0.f, 0.f);
#pragma unroll
    for (int i = 0; i < 8; ++i) { acc.x += sPart[i].x; acc.y += sPart[i].y; }
    partial[(size_t)blockIdx.y * gridDim.x + blockIdx.x] = acc;
  }
}

// Stage 2: 8 waves reduce the block partials through the matrix pipe.
// V_WMMA_F32_16X16X4_F32 with B = ones:  D[m][n] = sum_k A[m][k] + C[m][n].
// A 16x4 lane layout: lane L, vgpr v -> element (m = L%16, k = (L/16)*2 + v),
// so flat chunk index per lane/vgpr is m*4 + (L/16)*2 + v.  Adjacent chunk
// indices i0, i0+1 per lane -> one aligned float4 (b128) load of two float2
// partials.  REQUIRES count % 64 == 0 (guaranteed: count = 64 tiles * N).
// Every column of D holds identical row-sums; per-wave total =
// s(lane0) + s(lane16) where s = sum of the 8 D vgprs.
__global__ __launch_bounds__(256)
void pairwise_stage2(const float4* __restrict__ partial4,  // [count/2] float4
                     int count, float* __restrict__ out)
{
  const int lane   = threadIdx.x & 31;
  const int waveId = threadIdx.x >> 5;                 // 0..7
  const int idx0   = (lane & 15) * 4 + (lane >> 4) * 2; // even -> float4 aligned

  v2f ones; ones.x = 1.f; ones.y = 1.f;
  v8f accN = {}; v8f accD = {};

  // Waves stride the chunk space; loads from all 8 waves pipeline over L2 latency.
  for (int base = waveId * 64; base < count; base += 8 * 64) {
    const float4 pv = partial4[(base + idx0) >> 1];    // {n0,d0,n1,d1}
    v2f aN; aN.x = pv.x; aN.y = pv.z;
    v2f aD; aD.x = pv.y; aD.y = pv.w;
    accN = __builtin_amdgcn_wmma_f32_16x16x4_f32(false, aN, false, ones,
                                                 (short)0, accN, false, false);
    accD = __builtin_amdgcn_wmma_f32_16x16x4_f32(false, aD, false, ones,
                                                 (short)0, accD, false, false);
  }

  float sN = 0.f, sD = 0.f;
#pragma unroll
  for (int v = 0; v < 8; ++v) { sN += accN[v]; sD += accD[v]; }
  const float wN = __shfl(sN, 0, 32) + __shfl(sN, 16, 32);
  const float wD = __shfl(sD, 0, 32) + __shfl(sD, 16, 32);

  __shared__ float2 sWave[8];
  if (lane == 0) sWave[waveId] = make_float2(wN, wD);
  __syncthreads();
  if (threadIdx.x == 0) {
    float totN = 0.f, totD = 0.f;
#pragma unroll
    for (int i = 0; i < 8; ++i) { totN += sWave[i].x; totD += sWave[i].y; }
    out[0] = totN / fmaxf(totD, 1.0f);                 // LOSS_WEIGHT == 1
  }
}

extern "C" void kernel_launch(void* const* d_in, const int* in_sizes, int n_in,
                              void* d_out, int out_size, void* d_ws, size_t ws_size,
                              hipStream_t stream) {
  const float* ics  = (const float*)d_in[0];   // [N,8,256,256] f32
  const float* mask = (const float*)d_in[1];   // [N,256,256]   f32
  const float* gt   = (const float*)d_in[2];   // [N,256,256]   f32
  // d_in[3] = flag (unused, matches reference)
  float*  out     = (float*)d_out;
  float2* partial = (float2*)d_ws;             // 64 tiles * N * 8B = 32 KB for N=64

  const int N = in_sizes[1] / (HH * WW);
  const int tiles = (HH / TH) * (WW / TW);     // 64 blocks per image
  const int count = tiles * N;                 // multiple of 64 by construction

  dim3 grid(tiles, N);
  pairwise_stage1<<<grid, 256, 0, stream>>>(ics, mask, gt, partial);
  pairwise_stage2<<<1, 256, 0, stream>>>((const float4*)partial, count, out);
}